// MultiHeadAttention_7713761264335
// MI455X (gfx1250) — compile-verified
//
#include <hip/hip_runtime.h>

// ---------------------------------------------------------------------------
// Shapes
#define Bsz 16
#define Ssz 200
#define Hsz 128
#define NH  8
#define Dh  16          // head dim
#define SCALE 0.25f     // 1/sqrt(16)
#define EPS 1e-12f

typedef __attribute__((ext_vector_type(2))) float v2f;
typedef __attribute__((ext_vector_type(4))) float v4f;
typedef __attribute__((ext_vector_type(8))) float v8f;

// D = A(16x4) * B(4x16) + C, fp32 WMMA
__device__ __forceinline__ v8f wmma4(v2f a, v2f b, v8f c) {
    return __builtin_amdgcn_wmma_f32_16x16x4_f32(
        /*neg_a=*/false, a, /*neg_b=*/false, b,
        /*c_mod=*/(short)0, c, /*reuse_a=*/false, /*reuse_b=*/false);
}

// A-matrix 16x4 fp32 load: lanes 0-15 hold M=lane K={k,k+1}; lanes 16-31 K={k+2,k+3}
__device__ __forceinline__ v2f load_a(const float* A, int lda, int m0, int k) {
    int lane = threadIdx.x & 31;
    int r  = m0 + (lane & 15);
    int kk = k + ((lane >> 4) << 1);
    v2f a; a.x = A[r * lda + kk]; a.y = A[r * lda + kk + 1];
    return a;
}
// B-matrix 4x16 fp32 load (B[k][n]): lane holds column n = lane&15
__device__ __forceinline__ v2f load_b(const float* Bm, int ldb, int n0, int k) {
    int lane = threadIdx.x & 31;
    int c  = n0 + (lane & 15);
    int kk = k + ((lane >> 4) << 1);
    v2f b; b.x = Bm[kk * ldb + c]; b.y = Bm[(kk + 1) * ldb + c];
    return b;
}

// ---------------------------------------------------------------------------
// K1: q = x@Wq+bq ; k' = x@Wk+bk+pk ; v' = x@Wv+bv+pv   (layout [B*S, H])
__global__ void qkv_kernel(const float* __restrict__ x,
                           const float* __restrict__ Wq, const float* __restrict__ bq,
                           const float* __restrict__ Wk, const float* __restrict__ bk,
                           const float* __restrict__ Wv, const float* __restrict__ bv,
                           const float* __restrict__ pk, const float* __restrict__ pv,
                           float* __restrict__ q, float* __restrict__ k, float* __restrict__ v) {
    int tile = blockIdx.x;
    int n0 = (tile & 7) * 16;       // 8 N tiles over H=128
    int m0 = (tile >> 3) * 16;      // 200 M tiles over B*S=3200
    v8f cq = {}, ck = {}, cv = {};
    for (int kk = 0; kk < Hsz; kk += 4) {
        v2f a  = load_a(x, Hsz, m0, kk);
        v2f b0 = load_b(Wq, Hsz, n0, kk);
        v2f b1 = load_b(Wk, Hsz, n0, kk);
        v2f b2 = load_b(Wv, Hsz, n0, kk);
        cq = wmma4(a, b0, cq);
        ck = wmma4(a, b1, ck);
        cv = wmma4(a, b2, cv);
    }
    int lane = threadIdx.x & 31;
    int n  = n0 + (lane & 15);
    int mb = m0 + ((lane >> 4) << 3);
    float biasq = bq[n], biask = bk[n], biasv = bv[n];
    #pragma unroll
    for (int r = 0; r < 8; ++r) {
        int m = mb + r;
        q[m * Hsz + n] = cq[r] + biasq;
        k[m * Hsz + n] = ck[r] + biask + pk[m * Hsz + n];
        v[m * Hsz + n] = cv[r] + biasv + pv[m * Hsz + n];
    }
}

// ---------------------------------------------------------------------------
// K2: scores[b,h,i,j] = q_i . k'_j   (per (b,h): [S x 16] x [16 x S])
#define STILE 13   // ceil(200/16)
__global__ void scores_kernel(const float* __restrict__ q, const float* __restrict__ kp,
                              float* __restrict__ scores) {
    int idx = blockIdx.x;
    int jt = idx % STILE; idx /= STILE;
    int it = idx % STILE; idx /= STILE;
    int h = idx & 7, b = idx >> 3;
    int i0 = it * 16, j0 = jt * 16;
    const float* qb = q  + (size_t)(b * Ssz) * Hsz + h * Dh;   // row i: qb[i*H + d]
    const float* kb = kp + (size_t)(b * Ssz) * Hsz + h * Dh;   // row j: kb[j*H + d]
    int lane = threadIdx.x & 31;
    v8f c = {};
    #pragma unroll
    for (int kk = 0; kk < Dh; kk += 4) {
        int i = i0 + (lane & 15); if (i > Ssz - 1) i = Ssz - 1;   // clamp (stores bounded)
        int j = j0 + (lane & 15); if (j > Ssz - 1) j = Ssz - 1;
        int d = kk + ((lane >> 4) << 1);
        v2f a;  a.x = qb[i * Hsz + d]; a.y = qb[i * Hsz + d + 1];
        v2f bv; bv.x = kb[j * Hsz + d]; bv.y = kb[j * Hsz + d + 1];
        c = wmma4(a, bv, c);
    }
    float* srow = scores + (size_t)((b * NH + h) * Ssz) * Ssz;
    int j  = j0 + (lane & 15);
    int ib = i0 + ((lane >> 4) << 3);
    if (j < Ssz) {
        #pragma unroll
        for (int r = 0; r < 8; ++r) {
            int i = ib + r;
            if (i < Ssz) srow[(size_t)i * Ssz + j] = c[r];
        }
    }
}

// ---------------------------------------------------------------------------
// K3: s += tk[b,i,j,h].q_i ; s = s*SCALE + mask ; softmax over j (in-place)
// tk is streamed once (328 MB) -> non-temporal 128-bit loads.
__global__ void softmax_kernel(const float* __restrict__ q, const float* __restrict__ tk,
                               const float* __restrict__ mask, float* __restrict__ probs) {
    int row = blockIdx.x;                 // (b*NH+h)*S + i
    int i = row % Ssz; int bh = row / Ssz;
    int h = bh & 7, b = bh >> 3;
    int t = threadIdx.x;
    __shared__ float red[256];
    __shared__ __align__(16) float qsh[Dh];
    if (t < Dh) qsh[t] = q[(size_t)(b * Ssz + i) * Hsz + h * Dh + t];
    __syncthreads();
    bool act = t < Ssz;
    float sv = 0.f, s = -INFINITY;
    if (act) {
        const v4f* tkp = (const v4f*)(tk + ((size_t)(b * Ssz + i) * Ssz + t) * Hsz + h * Dh);
        const v4f* q4 = (const v4f*)qsh;
        v4f t0 = __builtin_nontemporal_load(tkp + 0);
        v4f t1 = __builtin_nontemporal_load(tkp + 1);
        v4f t2 = __builtin_nontemporal_load(tkp + 2);
        v4f t3 = __builtin_nontemporal_load(tkp + 3);
        v4f qa = q4[0], qb_ = q4[1], qc = q4[2], qd = q4[3];
        v4f acc = qa * t0 + qb_ * t1 + qc * t2 + qd * t3;
        float dot = probs[(size_t)row * Ssz + t] + acc.x + acc.y + acc.z + acc.w;
        sv = dot * SCALE + mask[((size_t)b * Ssz + i) * Ssz + t];
        s = sv;
    }
    red[t] = s; __syncthreads();
    for (int off = 128; off > 0; off >>= 1) {
        if (t < off) red[t] = fmaxf(red[t], red[t + off]);
        __syncthreads();
    }
    float m = red[0]; __syncthreads();
    float e = act ? expf(sv - m) : 0.f;
    red[t] = e; __syncthreads();
    for (int off = 128; off > 0; off >>= 1) {
        if (t < off) red[t] += red[t + off];
        __syncthreads();
    }
    float inv = 1.f / red[0];
    if (act) probs[(size_t)row * Ssz + t] = e * inv;
}

// ---------------------------------------------------------------------------
// K4: ctx[b,i,h,:] = sum_j probs[b,h,i,j] * v'[b,j,h,:]   (per (b,h): [SxS]x[Sx16])
__global__ void ctx1_kernel(const float* __restrict__ probs, const float* __restrict__ vp,
                            float* __restrict__ ctx) {
    int idx = blockIdx.x;
    int it = idx % STILE; idx /= STILE;
    int h = idx & 7, b = idx >> 3;
    int i0 = it * 16;
    const float* prow = probs + (size_t)((b * NH + h) * Ssz) * Ssz;
    const float* vb = vp + (size_t)(b * Ssz) * Hsz + h * Dh;
    int lane = threadIdx.x & 31;
    v8f c = {};
    for (int kk = 0; kk < Ssz; kk += 4) {       // 200 divisible by 4
        int i = i0 + (lane & 15); if (i > Ssz - 1) i = Ssz - 1;
        int j = kk + ((lane >> 4) << 1);
        int d = lane & 15;
        v2f a;  a.x = prow[(size_t)i * Ssz + j]; a.y = prow[(size_t)i * Ssz + j + 1];
        v2f bv; bv.x = vb[j * Hsz + d];          bv.y = vb[(j + 1) * Hsz + d];
        c = wmma4(a, bv, c);
    }
    int d  = lane & 15;
    int ib = i0 + ((lane >> 4) << 3);
    float* cb = ctx + (size_t)(b * Ssz) * Hsz + h * Dh;
    #pragma unroll
    for (int r = 0; r < 8; ++r) {
        int i = ib + r;
        if (i < Ssz) cb[i * Hsz + d] = c[r];
    }
}

// ---------------------------------------------------------------------------
// K5: ctx[b,i,h,d] += sum_j probs[b,h,i,j] * tv[b,i,j,h*16+d]  (wave per (b,h,i))
// One j per lane, 4x b128 non-temporal loads per lane; LDS-transpose reduction.
__global__ void ctx2_kernel(const float* __restrict__ probs, const float* __restrict__ tv,
                            float* __restrict__ ctx) {
    __shared__ float sh[8][32 * 17];
    int wid  = threadIdx.x >> 5;
    int lane = threadIdx.x & 31;
    int w = blockIdx.x * 8 + wid;                // global wave id = (b*NH+h)*S + i
    int i = w % Ssz; int bh = w / Ssz;
    int h = bh & 7, b = bh >> 3;
    const float* prow = probs + ((size_t)bh * Ssz + i) * Ssz;
    const v4f* tvb = (const v4f*)(tv + ((size_t)(b * Ssz + i) * Ssz) * Hsz + h * Dh);
    v4f a0 = {0.f, 0.f, 0.f, 0.f}, a1 = a0, a2 = a0, a3 = a0;
    #pragma unroll
    for (int jj = 0; jj < 7; ++jj) {             // 7*32 = 224 >= 200
        int j = jj * 32 + lane;
        if (j < Ssz) {
            float p = prow[j];
            const v4f* tj = tvb + (size_t)j * (Hsz / 4);
            v4f t0 = __builtin_nontemporal_load(tj + 0);
            v4f t1 = __builtin_nontemporal_load(tj + 1);
            v4f t2 = __builtin_nontemporal_load(tj + 2);
            v4f t3 = __builtin_nontemporal_load(tj + 3);
            a0 += p * t0; a1 += p * t1; a2 += p * t2; a3 += p * t3;
        }
    }
    float* rowp = &sh[wid][lane * 17];
    rowp[0]  = a0.x; rowp[1]  = a0.y; rowp[2]  = a0.z; rowp[3]  = a0.w;
    rowp[4]  = a1.x; rowp[5]  = a1.y; rowp[6]  = a1.z; rowp[7]  = a1.w;
    rowp[8]  = a2.x; rowp[9]  = a2.y; rowp[10] = a2.z; rowp[11] = a2.w;
    rowp[12] = a3.x; rowp[13] = a3.y; rowp[14] = a3.z; rowp[15] = a3.w;
    __syncthreads();
    // lane l: component d = l&15, sums rows [r0, r0+16)
    int d = lane & 15, r0 = (lane >> 4) << 4;
    float s = 0.f;
    #pragma unroll
    for (int r = 0; r < 16; ++r) s += sh[wid][(r0 + r) * 17 + d];
    s += __shfl_xor(s, 16, 32);
    if (lane < 16) {
        float* cp = ctx + (size_t)(b * Ssz + i) * Hsz + h * Dh + d;
        *cp += s;
    }
}

// ---------------------------------------------------------------------------
// K6a: x = ctx @ Wd + bd + input
__global__ void outproj_kernel(const float* __restrict__ ctx, const float* __restrict__ Wd,
                               const float* __restrict__ bd, const float* __restrict__ inp,
                               float* __restrict__ xout) {
    int tile = blockIdx.x;
    int n0 = (tile & 7) * 16;
    int m0 = (tile >> 3) * 16;
    v8f c = {};
    for (int kk = 0; kk < Hsz; kk += 4) {
        v2f a = load_a(ctx, Hsz, m0, kk);
        v2f b = load_b(Wd, Hsz, n0, kk);
        c = wmma4(a, b, c);
    }
    int lane = threadIdx.x & 31;
    int n  = n0 + (lane & 15);
    int mb = m0 + ((lane >> 4) << 3);
    float bias = bd[n];
    #pragma unroll
    for (int r = 0; r < 8; ++r) {
        int m = mb + r;
        xout[m * Hsz + n] = c[r] + bias + inp[m * Hsz + n];
    }
}

// K6b: LayerNorm per row of 128
__global__ void ln_kernel(const float* __restrict__ x, const float* __restrict__ gamma,
                          const float* __restrict__ beta, float* __restrict__ out) {
    int row = blockIdx.x, t = threadIdx.x;
    __shared__ float red[128];
    float v = x[(size_t)row * Hsz + t];
    red[t] = v; __syncthreads();
    for (int off = 64; off > 0; off >>= 1) {
        if (t < off) red[t] += red[t + off];
        __syncthreads();
    }
    float mu = red[0] * (1.f / Hsz); __syncthreads();
    float dv = v - mu;
    red[t] = dv * dv; __syncthreads();
    for (int off = 64; off > 0; off >>= 1) {
        if (t < off) red[t] += red[t + off];
        __syncthreads();
    }
    float var = red[0] * (1.f / Hsz);
    out[(size_t)row * Hsz + t] = gamma[t] * dv * rsqrtf(var + EPS) + beta[t];
}

// ---------------------------------------------------------------------------
extern "C" void kernel_launch(void* const* d_in, const int* in_sizes, int n_in,
                              void* d_out, int out_size, void* d_ws, size_t ws_size,
                              hipStream_t stream) {
    const float* x    = (const float*)d_in[0];
    const float* mask = (const float*)d_in[1];
    const float* tk   = (const float*)d_in[2];
    const float* tv   = (const float*)d_in[3];
    const float* pk   = (const float*)d_in[4];
    const float* pv   = (const float*)d_in[5];
    const float* Wq = (const float*)d_in[6];  const float* bq = (const float*)d_in[7];
    const float* Wk = (const float*)d_in[8];  const float* bk = (const float*)d_in[9];
    const float* Wv = (const float*)d_in[10]; const float* bv = (const float*)d_in[11];
    const float* Wd = (const float*)d_in[12]; const float* bd = (const float*)d_in[13];
    const float* gamma = (const float*)d_in[14];
    const float* beta  = (const float*)d_in[15];
    float* out = (float*)d_out;

    float* ws = (float*)d_ws;
    const size_t NQ = (size_t)Bsz * Ssz * Hsz;         // 409600
    float* q  = ws;
    float* kp = ws + NQ;
    float* vp = ws + 2 * NQ;
    float* pr = ws + 3 * NQ;                           // scores -> probs (5.12M)
    float* ct = ws + 3 * NQ + (size_t)Bsz * NH * Ssz * Ssz;
    float* xr = ct + NQ;

    qkv_kernel    <<<(Bsz * Ssz / 16) * (Hsz / 16), 32, 0, stream>>>(x, Wq, bq, Wk, bk, Wv, bv, pk, pv, q, kp, vp);
    scores_kernel <<<Bsz * NH * STILE * STILE, 32, 0, stream>>>(q, kp, pr);
    softmax_kernel<<<Bsz * NH * Ssz, 256, 0, stream>>>(q, tk, mask, pr);
    ctx1_kernel   <<<Bsz * NH * STILE, 32, 0, stream>>>(pr, vp, ct);
    ctx2_kernel   <<<Bsz * NH * Ssz / 8, 256, 0, stream>>>(pr, tv, ct);
    outproj_kernel<<<(Bsz * Ssz / 16) * (Hsz / 16), 32, 0, stream>>>(ct, Wd, bd, x, xr);
    ln_kernel     <<<Bsz * Ssz, 128, 0, stream>>>(xr, gamma, beta, out);
}